// yoloLoss_16698832847214
// MI455X (gfx1250) — compile-verified
//
#include <hip/hip_runtime.h>
#include <stdint.h>

// ---------------- problem constants (from the reference) ----------------
// N=64, S=112, CEIL=25, T = 64*112*112 = 802816
// object cells: flat%8 in {0,1}  -> pairs (8k, 8k+1), k in [0, T/8)
#define NPAIRS        100352          // T/8
#define PAIRS_PER_BLK 128
#define THREADS       128             // 4 waves of 32
#define NBLOCKS       (NPAIRS / PAIRS_PER_BLK)   // 784
#define ROW_F         50              // floats per pair row in LDS (stride 50 vs 64 banks -> conflict-free)
#define INV_N         (1.0f / 64.0f)

typedef __attribute__((ext_vector_type(2))) float v2f;
typedef __attribute__((ext_vector_type(8))) float v8f;

#ifndef __has_builtin
#define __has_builtin(x) 0
#endif

#if defined(__gfx1250__) && __has_builtin(__builtin_amdgcn_global_load_async_to_lds_b64)
#define ASYNC_OK 1
#else
#define ASYNC_OK 0
#endif

#if defined(__gfx1250__) && __has_builtin(__builtin_amdgcn_wmma_f32_16x16x4_f32)
#define WMMA_OK 1
#else
#define WMMA_OK 0
#endif

#if ASYNC_OK
// builtin prototype (from clang diagnostic): param0 = v2i in addrspace(1) ("__device__"),
// param1 = v2i in addrspace(3) ("__shared__"), then imm offset, imm cpol.
typedef int v2i __attribute__((vector_size(8)));
typedef __attribute__((address_space(1))) v2i* g64p;
typedef __attribute__((address_space(3))) v2i* l64p;

__device__ __forceinline__ void async_copy_b64(const void* gsrc, void* ldst) {
    __builtin_amdgcn_global_load_async_to_lds_b64(
        (g64p)(v2i*)gsrc, (l64p)(v2i*)ldst, 0, 0);
}
#endif

__device__ __forceinline__ float sigf(float x) { return 1.0f / (1.0f + __expf(-x)); }

__global__ __launch_bounds__(THREADS) void yolo_pair_kernel(
    const float* __restrict__ pred, const float* __restrict__ tgt,
    float* __restrict__ blockSums)
{
    __shared__ float ldsP[PAIRS_PER_BLK * ROW_F];   // 25.6 KB
    __shared__ float ldsT[PAIRS_PER_BLK * ROW_F];   // 25.6 KB
    __shared__ float wred[THREADS / 32];

    const int tid = threadIdx.x;
    const int blk = blockIdx.x;
    // byte offset of this block's first pair record (stride 800 B, payload 200 B)
    const long long baseByte = (long long)blk * PAIRS_PER_BLK * 800ll;

    // ---- stage 128 pairs x 50 floats x 2 tensors into LDS ----
    // 128*25 = 3200 qwords per tensor; 25 iterations per thread.
#if ASYNC_OK
    {
        const char* gP = (const char*)pred + baseByte;
        const char* gT = (const char*)tgt  + baseByte;
        for (int q = tid; q < PAIRS_PER_BLK * 25; q += THREADS) {
            const int pr = q / 25, ch = q % 25;
            const int gOff = pr * 800 + ch * 8;           // strided global record
            const int lOff = pr * (ROW_F * 4) + ch * 8;   // packed LDS row
            async_copy_b64(gP + gOff, (char*)ldsP + lOff);
            async_copy_b64(gT + gOff, (char*)ldsT + lOff);
        }
#if __has_builtin(__builtin_amdgcn_s_wait_asynccnt)
        __builtin_amdgcn_s_wait_asynccnt(0);
#else
        asm volatile("s_wait_asynccnt 0" ::: "memory");
#endif
    }
#else
    {
        const float2* gP = (const float2*)((const char*)pred + baseByte);
        const float2* gT = (const float2*)((const char*)tgt  + baseByte);
        for (int q = tid; q < PAIRS_PER_BLK * 25; q += THREADS) {
            const int pr = q / 25, ch = q % 25;
            ((float2*)(ldsP + pr * ROW_F))[ch] = gP[pr * 100 + ch]; // 100 float2 per 800B record
            ((float2*)(ldsT + pr * ROW_F))[ch] = gT[pr * 100 + ch];
        }
    }
#endif
    __syncthreads();

    // ---- one pair per thread (uniform control flow: EXEC all ones for WMMA) ----
    const float* p = ldsP + tid * ROW_F;   // [0..24] cell0, [25..49] cell1
    const float* t = ldsT + tid * ROW_F;

    float acc = 0.0f;

    // class loss (channels 5..24, both cells of the pair)
#pragma unroll
    for (int ch = 5; ch < 25; ++ch) {
        const float d0 = p[ch]      - t[ch];
        const float d1 = p[25 + ch] - t[25 + ch];
        acc += d0 * d0 + d1 * d1;
    }

    // boxes
    float bp[2][5], bt[2][5];
#pragma unroll
    for (int i = 0; i < 5; ++i) {
        bp[0][i] = p[i];  bp[1][i] = p[25 + i];
        bt[0][i] = t[i];  bt[1][i] = t[25 + i];
    }

    // target box = FIRST box of the pair (as in the reference)
    const float b2x1 = bt[0][0] - 0.5f * bt[0][2], b2y1 = bt[0][1] - 0.5f * bt[0][3];
    const float b2x2 = bt[0][0] + 0.5f * bt[0][2], b2y2 = bt[0][1] + 0.5f * bt[0][3];
    const float a2 = (b2x2 - b2x1) * (b2y2 - b2y1);

    float iou[2];
#pragma unroll
    for (int j = 0; j < 2; ++j) {
        const float x1 = bp[j][0] - 0.5f * bp[j][2], y1 = bp[j][1] - 0.5f * bp[j][3];
        const float x2 = bp[j][0] + 0.5f * bp[j][2], y2 = bp[j][1] + 0.5f * bp[j][3];
        const float wx = fmaxf(fminf(x2, b2x2) - fmaxf(x1, b2x1), 0.0f);
        const float wy = fmaxf(fminf(y2, b2y2) - fmaxf(y1, b2y1), 0.0f);
        const float inter = wx * wy;
        const float a1 = (x2 - x1) * (y2 - y1);
        iou[j] = inter / (a1 + a2 - inter);
    }
    const int   mi   = (iou[1] > iou[0]) ? 1 : 0;    // argmax: first index on tie
    const float miou = fmaxf(iou[0], iou[1]);        // detached in reference

    const float* pr5 = bp[mi];
    const float* tr5 = bt[mi];

    const float dc = sigf(pr5[4]) - miou;
    const float dx = sigf(pr5[0]) - tr5[0];
    const float dy = sigf(pr5[1]) - tr5[1];
    const float ew = __expf(pr5[2]) - __expf(tr5[2]);
    const float eh = __expf(pr5[3]) - __expf(tr5[3]);

    acc += 5.0f * (dx * dx + dy * dy + ew * ew + eh * eh) + dc * dc;

    // ---- wave reduction ----
    float wsum;
#if WMMA_OK
    // A(16x4 f32): lane m holds {A[m][0],A[m][1]}, lane m+16 holds {A[m][2],A[m][3]}.
    // With a = {acc, 0} and B = ones:  D[m][n] = acc(lane m) + acc(lane m+16).
    // D layout: c[r] = D[r + (lane>=16 ? 8 : 0)][*]  ->  sum_r c[r] is half-wave row sum;
    // adding the xor-16 partner yields the full 32-lane sum on every lane.
    v2f a; a.x = acc;  a.y = 0.0f;
    v2f b; b.x = 1.0f; b.y = 1.0f;
    v8f c = {};
    c = __builtin_amdgcn_wmma_f32_16x16x4_f32(false, a, false, b, (short)0, c, false, false);
    float s = c[0] + c[1] + c[2] + c[3] + c[4] + c[5] + c[6] + c[7];
    wsum = s + __shfl_xor(s, 16, 32);
#else
    float s = acc;
#pragma unroll
    for (int o = 16; o > 0; o >>= 1) s += __shfl_xor(s, o, 32);
    wsum = s;
#endif

    if ((tid & 31) == 0) wred[tid >> 5] = wsum;
    __syncthreads();
    if (tid == 0) {
        float bs = 0.0f;
#pragma unroll
        for (int w = 0; w < THREADS / 32; ++w) bs += wred[w];
        blockSums[blk] = bs;                       // deterministic per-block partial
    }
}

__global__ __launch_bounds__(256) void yolo_finalize_kernel(
    const float* __restrict__ blockSums, float* __restrict__ out)
{
    __shared__ float sm[256];
    const int tid = threadIdx.x;
    float s = 0.0f;
    for (int i = tid; i < NBLOCKS; i += 256) s += blockSums[i];   // fixed order
    sm[tid] = s;
    __syncthreads();
    for (int w = 128; w > 0; w >>= 1) {
        if (tid < w) sm[tid] += sm[tid + w];
        __syncthreads();
    }
    if (tid == 0) out[0] = sm[0] * INV_N;
}

extern "C" void kernel_launch(void* const* d_in, const int* in_sizes, int n_in,
                              void* d_out, int out_size, void* d_ws, size_t ws_size,
                              hipStream_t stream) {
    (void)in_sizes; (void)n_in; (void)out_size; (void)ws_size;
    const float* pred = (const float*)d_in[0];
    const float* tgt  = (const float*)d_in[1];
    float* blockSums  = (float*)d_ws;     // NBLOCKS floats
    float* out        = (float*)d_out;

    yolo_pair_kernel<<<NBLOCKS, THREADS, 0, stream>>>(pred, tgt, blockSums);
    yolo_finalize_kernel<<<1, 256, 0, stream>>>(blockSums, out);
}